// GCN_43430709297957
// MI455X (gfx1250) — compile-verified
//
#include <hip/hip_runtime.h>
#include <hip/hip_bf16.h>

typedef __attribute__((ext_vector_type(2))) float v2f;
typedef __attribute__((ext_vector_type(8))) float v8f;

#define N_NODES 50000
#define N_FEAT  128
#define HIDDEN  64
#define N_EDGES 800000
#define M_TILES (N_NODES / 16)   // 3125 exactly

// ---------------- degree / normalization ----------------

__global__ void gcn_init_deg(float* __restrict__ deg) {
    int i = blockIdx.x * blockDim.x + threadIdx.x;
    if (i < N_NODES) deg[i] = 1.0f;   // self-loop
}

__global__ void gcn_add_deg(const int* __restrict__ dst, float* __restrict__ deg) {
    int e = blockIdx.x * blockDim.x + threadIdx.x;
    if (e < N_EDGES) atomicAdd(&deg[dst[e]], 1.0f);
}

__global__ void gcn_rsqrt_deg(float* __restrict__ deg) {
    int i = blockIdx.x * blockDim.x + threadIdx.x;
    if (i < N_NODES) deg[i] = rsqrtf(deg[i]);
}

__global__ void gcn_zero(float* __restrict__ p, int n) {
    int i = blockIdx.x * blockDim.x + threadIdx.x;
    if (i < n) p[i] = 0.0f;
}

// ---------------- WMMA GEMM: H[M,64] = A[M,K] * W[K,64] ----------------
// One wave per 16-row M tile; 4 fp32 16x16 accumulators cover all 64 cols.
template <int K>
__global__ void gcn_gemm_wmma(const float* __restrict__ A,
                              const float* __restrict__ W,
                              float* __restrict__ H) {
    const int wave = (blockIdx.x * blockDim.x + threadIdx.x) >> 5;
    const int lane = threadIdx.x & 31;
    if (wave >= M_TILES) return;                 // wave-uniform exit, EXEC stays all-1s

    const int m0   = wave * 16;
    const int mrow = m0 + (lane & 15);           // A-frag: lanes 0-15 hold rows M=0..15
    const int kh   = (lane >> 4) << 1;           // upper half-wave holds K+2,K+3
    const int ncol = lane & 15;                  // B-frag: lanes 0-15 hold cols N=0..15

    v8f acc0 = {}, acc1 = {}, acc2 = {}, acc3 = {};
    const float* __restrict__ arow = A + (size_t)mrow * K;

#pragma unroll 4
    for (int k = 0; k < K; k += 4) {
        v2f a;
        a[0] = arow[k + kh];
        a[1] = arow[k + kh + 1];

        const float* __restrict__ w0 = W + (size_t)(k + kh) * HIDDEN;
        const float* __restrict__ w1 = W + (size_t)(k + kh + 1) * HIDDEN;
        v2f b0 = { w0[ncol],      w1[ncol]      };
        v2f b1 = { w0[16 + ncol], w1[16 + ncol] };
        v2f b2 = { w0[32 + ncol], w1[32 + ncol] };
        v2f b3 = { w0[48 + ncol], w1[48 + ncol] };

        acc0 = __builtin_amdgcn_wmma_f32_16x16x4_f32(false, a, false, b0, (short)0, acc0, false, false);
        acc1 = __builtin_amdgcn_wmma_f32_16x16x4_f32(false, a, false, b1, (short)0, acc1, false, false);
        acc2 = __builtin_amdgcn_wmma_f32_16x16x4_f32(false, a, false, b2, (short)0, acc2, false, false);
        acc3 = __builtin_amdgcn_wmma_f32_16x16x4_f32(false, a, false, b3, (short)0, acc3, false, false);
    }

    // C/D layout: VGPR v holds row m0 + v (+8 for upper half-wave), col = lane&15
    const int rbase = (lane >> 4) << 3;
#pragma unroll
    for (int v = 0; v < 8; ++v) {
        float* __restrict__ out = H + (size_t)(m0 + rbase + v) * HIDDEN;
        out[ncol]      = acc0[v];
        out[16 + ncol] = acc1[v];
        out[32 + ncol] = acc2[v];
        out[48 + ncol] = acc3[v];
    }
}

// ---------------- edge scatter-aggregate (L2-resident atomics) ----------------
// 16 threads per edge, float4 per thread; all atomics land in the 192MB L2.
__global__ void gcn_scatter(const float* __restrict__ h,
                            const int* __restrict__ src,
                            const int* __restrict__ dst,
                            const float* __restrict__ dinv,
                            float* __restrict__ agg) {
    long long t = (long long)blockIdx.x * blockDim.x + threadIdx.x;
    int e = (int)(t >> 4);
    if (e >= N_EDGES) return;
    int f = ((int)t & 15) << 2;

    int s = src[e];
    int d = dst[e];
    float c = dinv[s] * dinv[d];

    const float4 v = *(const float4*)(h + (size_t)s * HIDDEN + f);
    float* __restrict__ ap = agg + (size_t)d * HIDDEN + f;
    atomicAdd(ap + 0, v.x * c);
    atomicAdd(ap + 1, v.y * c);
    atomicAdd(ap + 2, v.z * c);
    atomicAdd(ap + 3, v.w * c);
}

// ---------------- epilogue: self-loop + bias + ReLU ----------------
__global__ void gcn_finalize(const float* __restrict__ agg,
                             const float* __restrict__ h,
                             const float* __restrict__ dinv,
                             const float* __restrict__ bias,
                             float* __restrict__ out) {
    int t = blockIdx.x * blockDim.x + threadIdx.x;
    if (t >= N_NODES * HIDDEN) return;
    int node = t >> 6;          // / HIDDEN
    int f    = t & (HIDDEN - 1);
    float di = dinv[node];
    float v  = agg[t] + h[t] * (di * di) + bias[f];
    out[t] = v > 0.0f ? v : 0.0f;
}

// ---------------- launcher ----------------

extern "C" void kernel_launch(void* const* d_in, const int* in_sizes, int n_in,
                              void* d_out, int out_size, void* d_ws, size_t ws_size,
                              hipStream_t stream) {
    const float* x  = (const float*)d_in[0];
    const int*   ei = (const int*)d_in[1];       // [2, E] row-major int32
    const float* W1 = (const float*)d_in[2];
    const float* b1 = (const float*)d_in[3];
    const float* W2 = (const float*)d_in[4];
    const float* b2 = (const float*)d_in[5];
    const int* src = ei;
    const int* dst = ei + N_EDGES;
    float* out = (float*)d_out;

    const int NH = N_NODES * HIDDEN;             // 3.2M floats
    float* dinv = (float*)d_ws;                  // [50176] (padded to 256)
    float* h    = dinv + 50176;                  // [NH]  GEMM result
    float* agg  = h + NH;                        // [NH]  aggregation buffer
    float* y    = agg + NH;                      // [NH]  layer-1 activations

    const int T256n  = (N_NODES + 255) / 256;
    const int T256e  = (N_EDGES + 255) / 256;
    const int T256nh = (NH + 255) / 256;
    const int Tgemm  = (M_TILES * 32 + 255) / 256;
    const int Tscat  = (int)(((long long)N_EDGES * 16 + 255) / 256);

    // normalization coefficients (shared by both layers)
    gcn_init_deg<<<T256n, 256, 0, stream>>>(dinv);
    gcn_add_deg<<<T256e, 256, 0, stream>>>(dst, dinv);
    gcn_rsqrt_deg<<<T256n, 256, 0, stream>>>(dinv);

    // ---- layer 1 ----
    gcn_gemm_wmma<N_FEAT><<<Tgemm, 256, 0, stream>>>(x, W1, h);
    gcn_zero<<<T256nh, 256, 0, stream>>>(agg, NH);
    gcn_scatter<<<Tscat, 256, 0, stream>>>(h, src, dst, dinv, agg);
    gcn_finalize<<<T256nh, 256, 0, stream>>>(agg, h, dinv, b1, y);

    // ---- layer 2 ----
    gcn_gemm_wmma<HIDDEN><<<Tgemm, 256, 0, stream>>>(y, W2, h);
    gcn_zero<<<T256nh, 256, 0, stream>>>(agg, NH);
    gcn_scatter<<<Tscat, 256, 0, stream>>>(h, src, dst, dinv, agg);
    gcn_finalize<<<T256nh, 256, 0, stream>>>(agg, h, dinv, b2, out);
}